// MultiQueryAttention_6442450944648
// MI455X (gfx1250) — compile-verified
//
#include <hip/hip_runtime.h>
#include <hip/hip_bf16.h>

typedef __attribute__((ext_vector_type(2))) float v2f;
typedef __attribute__((ext_vector_type(8))) float v8f;

// ---------------------------------------------------------------------------
// C[M=1024, N=1024] = A[1024,1024] @ W[1024,1024] + bias[N]
// Block: 256 threads = 8 wave32; block tile 64(M) x 128(N); K-step 16.
// Each wave computes a 32x32 tile as a 2x2 grid of 16x16 WMMA accumulators,
// so every A-frag is reused across 2 N-subtiles and every B-frag across 2
// M-subtiles (2 LDS dwords per v_wmma_f32_16x16x4_f32).
// B panel is stored pair-interleaved in LDS ([k/2][n][2]) so a B-fragment is
// one aligned ds_load_b64 -> no VALU repacking before the WMMA.
// ---------------------------------------------------------------------------
__global__ __launch_bounds__(256) void gemm1024_bias_wmma(
    const float* __restrict__ A, const float* __restrict__ W,
    const float* __restrict__ bias, float* __restrict__ C) {
  constexpr int K = 1024, N = 1024;
  __shared__ __align__(16) float As[64 * 16];        // [m][k]       4 KB
  __shared__ __align__(16) float BsP[8 * 128 * 2];   // [k/2][n][2]  8 KB

  const int tid  = threadIdx.x;
  const int lane = tid & 31;
  const int wid  = tid >> 5;
  const int half = lane >> 4;   // 0: lanes 0-15, 1: lanes 16-31
  const int l16  = lane & 15;

  const int m0 = blockIdx.y * 64;
  const int n0 = blockIdx.x * 128;
  const int mw = (wid & 1) * 32;   // wave M offset in block tile
  const int nw = (wid >> 1) * 32;  // wave N offset in block tile

  // cooperative-load coordinates
  const int ar = tid >> 2;         // 0..63 : A panel row
  const int ac = (tid & 3) * 4;    // 0,4,8,12
  const int kp = tid >> 5;         // 0..7  : B panel k-pair
  const int n4 = (tid & 31) * 4;   // 0..124

  v8f acc00 = {}, acc01 = {}, acc10 = {}, acc11 = {};

  for (int k0 = 0; k0 < K; k0 += 16) {
    // A panel [64 x 16]
    const float4 av = *reinterpret_cast<const float4*>(&A[(size_t)(m0 + ar) * K + k0 + ac]);
    *reinterpret_cast<float4*>(&As[ar * 16 + ac]) = av;
    // B panel, pair-interleaved: {W[2kp, n], W[2kp+1, n]} contiguous
    const float4 f0 = *reinterpret_cast<const float4*>(&W[(size_t)(k0 + 2 * kp) * N + n0 + n4]);
    const float4 f1 = *reinterpret_cast<const float4*>(&W[(size_t)(k0 + 2 * kp + 1) * N + n0 + n4]);
    const int bbase = (kp * 128 + n4) * 2;
    float4 s0; s0.x = f0.x; s0.y = f1.x; s0.z = f0.y; s0.w = f1.y;
    float4 s1; s1.x = f0.z; s1.y = f1.z; s1.z = f0.w; s1.w = f1.w;
    *reinterpret_cast<float4*>(&BsP[bbase])     = s0;
    *reinterpret_cast<float4*>(&BsP[bbase + 4]) = s1;
    __syncthreads();

#pragma unroll
    for (int kk = 0; kk < 16; kk += 4) {
      const int kb = kk + 2 * half;  // even; A-frag lanes 0-15: K={kk,kk+1}, 16-31: K={kk+2,kk+3}
      const int kq = kb >> 1;
      v2f a0 = *reinterpret_cast<const v2f*>(&As[(mw + l16) * 16 + kb]);
      v2f a1 = *reinterpret_cast<const v2f*>(&As[(mw + 16 + l16) * 16 + kb]);
      v2f b0 = *reinterpret_cast<const v2f*>(&BsP[(kq * 128 + nw + l16) * 2]);
      v2f b1 = *reinterpret_cast<const v2f*>(&BsP[(kq * 128 + nw + 16 + l16) * 2]);
      acc00 = __builtin_amdgcn_wmma_f32_16x16x4_f32(false, a0, false, b0, (short)0, acc00, false, false);
      acc01 = __builtin_amdgcn_wmma_f32_16x16x4_f32(false, a0, false, b1, (short)0, acc01, false, false);
      acc10 = __builtin_amdgcn_wmma_f32_16x16x4_f32(false, a1, false, b0, (short)0, acc10, false, false);
      acc11 = __builtin_amdgcn_wmma_f32_16x16x4_f32(false, a1, false, b1, (short)0, acc11, false, false);
    }
    __syncthreads();
  }

  const float bias0 = bias[n0 + nw + l16];
  const float bias1 = bias[n0 + nw + 16 + l16];
#pragma unroll
  for (int r = 0; r < 8; ++r) {
    const int row0 = m0 + mw + r + 8 * half;       // C layout: vgpr r -> M=r / M=8+r per half
    const int row1 = row0 + 16;
    C[(size_t)row0 * N + n0 + nw + l16]      = acc00[r] + bias0;
    C[(size_t)row0 * N + n0 + nw + 16 + l16] = acc01[r] + bias1;
    C[(size_t)row1 * N + n0 + nw + l16]      = acc10[r] + bias0;
    C[(size_t)row1 * N + n0 + nw + 16 + l16] = acc11[r] + bias1;
  }
}

// ---------------------------------------------------------------------------
// Multi-query attention core. One wave32 per 16-row query tile.
//   Q: [B, 512, 1024] (per-head slice h*64..h*64+63)
//   K2/V2: [B, 8192, 64]  (k/v projections viewed as single head of len S*H)
//   Y: scrambled output, per batch flat index (h*64+d)*512 + s  (matches the
//      reference's swapaxes(-1,-2).reshape(B,-1,D))
// Flash-attention style online softmax; Q tile lives in VGPRs as WMMA A-frags.
// ---------------------------------------------------------------------------
__global__ __launch_bounds__(32) void mqa_attention_wmma(
    const float* __restrict__ Q, const float* __restrict__ Kp,
    const float* __restrict__ Vp, float* __restrict__ Y) {
  constexpr int S = 512, HD = 64, SK = 8192;
  const int sTile = blockIdx.x;  // 0..31
  const int h     = blockIdx.y;  // 0..15
  const int b     = blockIdx.z;  // 0..1
  const int lane  = threadIdx.x & 31;
  const int half  = lane >> 4;
  const int l16   = lane & 15;
  const int sBase = sTile * 16;

  __shared__ __align__(8) float p_lds[16 * 16];

  // Preload Q tile [16 x 64] as 16 A-fragments (K-steps of 4): 32 VGPRs/lane.
  const float* qrow = Q + ((size_t)b * S + sBase + l16) * 1024 + h * 64 + 2 * half;
  v2f qa[16];
#pragma unroll
  for (int kk = 0; kk < 16; ++kk)
    qa[kk] = *reinterpret_cast<const v2f*>(qrow + kk * 4);

  const float* kbase = Kp + (size_t)b * SK * HD;
  const float* vbase = Vp + (size_t)b * SK * HD;

  float mrow[8], lrow[8];
#pragma unroll
  for (int r = 0; r < 8; ++r) { mrow[r] = -__builtin_inff(); lrow[r] = 0.0f; }
  v8f accO[4] = {{}, {}, {}, {}};

  for (int j0 = 0; j0 < SK; j0 += 16) {
    if (j0 + 16 < SK) {
      __builtin_prefetch(kbase + (size_t)(j0 + 16 + l16) * HD, 0, 0);
      __builtin_prefetch(vbase + (size_t)(j0 + 16 + l16) * HD, 0, 0);
    }

    // scores tile S16 = Q[16x64] @ K2[j0:j0+16, :]^T  (B-frag: n=j lane, k=d)
    v8f sc = {};
    const float* krow = kbase + (size_t)(j0 + l16) * HD + 2 * half;
#pragma unroll
    for (int kk = 0; kk < 16; ++kk) {
      v2f bf = *reinterpret_cast<const v2f*>(krow + kk * 4);
      sc = __builtin_amdgcn_wmma_f32_16x16x4_f32(false, qa[kk], false, bf, (short)0, sc, false, false);
    }

    // online softmax (rows split across wave halves exactly like the C layout)
#pragma unroll
    for (int r = 0; r < 8; ++r) {
      float c = sc[r] * 0.125f;  // 1/sqrt(64)
      float rmax = c;
      rmax = fmaxf(rmax, __shfl_xor(rmax, 1, 32));
      rmax = fmaxf(rmax, __shfl_xor(rmax, 2, 32));
      rmax = fmaxf(rmax, __shfl_xor(rmax, 4, 32));
      rmax = fmaxf(rmax, __shfl_xor(rmax, 8, 32));
      const float mnew = fmaxf(mrow[r], rmax);
      const float p = __expf(c - mnew);
      float ps = p;
      ps += __shfl_xor(ps, 1, 32);
      ps += __shfl_xor(ps, 2, 32);
      ps += __shfl_xor(ps, 4, 32);
      ps += __shfl_xor(ps, 8, 32);
      const float scl = __expf(mrow[r] - mnew);
      lrow[r] = lrow[r] * scl + ps;
      mrow[r] = mnew;
#pragma unroll
      for (int t = 0; t < 4; ++t) accO[t][r] *= scl;
      p_lds[(r + 8 * half) * 16 + l16] = p;  // stage P for A-frag re-layout
    }
    __syncthreads();  // single-wave WG: cheap; orders LDS write -> read

    // accO[16x64] += P[16x16] @ V2[j0:j0+16, 0:64]
#pragma unroll
    for (int kk = 0; kk < 4; ++kk) {
      const int kb = 4 * kk + 2 * half;
      v2f pa = *reinterpret_cast<const v2f*>(&p_lds[l16 * 16 + kb]);
      const float* vrow = vbase + (size_t)(j0 + kb) * HD;
#pragma unroll
      for (int t = 0; t < 4; ++t) {
        v2f bf;
        bf.x = vrow[t * 16 + l16];
        bf.y = vrow[HD + t * 16 + l16];
        accO[t] = __builtin_amdgcn_wmma_f32_16x16x4_f32(false, pa, false, bf, (short)0, accO[t], false, false);
      }
    }
    __syncthreads();
  }

  // normalize and write scrambled layout: Y[b].flat[(h*64+d)*512 + s]
  float* ybase = Y + (size_t)b * (S * 1024);
#pragma unroll
  for (int r = 0; r < 8; ++r) {
    const int sIdx = sBase + r + 8 * half;
    const float inv = 1.0f / lrow[r];
#pragma unroll
    for (int t = 0; t < 4; ++t) {
      const int d = t * 16 + l16;
      ybase[(size_t)(h * 64 + d) * 512 + sIdx] = accO[t][r] * inv;
    }
  }
}

// ---------------------------------------------------------------------------
// Launch: 3 projection GEMMs -> attention -> output GEMM (consumes scrambled Y
// as a plain row-major [1024,1024] matrix, matching the reference faithfully).
// Workspace: Q(4MB) | K(4MB) | V(4MB) | Y(4MB) = 16MB.
// ---------------------------------------------------------------------------
extern "C" void kernel_launch(void* const* d_in, const int* in_sizes, int n_in,
                              void* d_out, int out_size, void* d_ws, size_t ws_size,
                              hipStream_t stream) {
  (void)in_sizes; (void)n_in; (void)out_size; (void)ws_size;
  const float* x  = (const float*)d_in[0];
  const float* Wq = (const float*)d_in[1];
  const float* bq = (const float*)d_in[2];
  const float* Wk = (const float*)d_in[3];
  const float* bk = (const float*)d_in[4];
  const float* Wv = (const float*)d_in[5];
  const float* bv = (const float*)d_in[6];
  const float* Wo = (const float*)d_in[7];
  const float* bo = (const float*)d_in[8];
  float* out = (float*)d_out;

  float* ws = (float*)d_ws;
  float* Qp = ws;
  float* Kp = ws + (1u << 20);
  float* Vp = ws + (2u << 20);
  float* Yp = ws + (3u << 20);

  const dim3 gGrid(8, 16, 1);   // N tiles of 128, M tiles of 64
  const dim3 gBlk(256, 1, 1);
  gemm1024_bias_wmma<<<gGrid, gBlk, 0, stream>>>(x, Wq, bq, Qp);
  gemm1024_bias_wmma<<<gGrid, gBlk, 0, stream>>>(x, Wk, bk, Kp);
  gemm1024_bias_wmma<<<gGrid, gBlk, 0, stream>>>(x, Wv, bv, Vp);
  mqa_attention_wmma<<<dim3(32, 16, 2), dim3(32, 1, 1), 0, stream>>>(Qp, Kp, Vp, Yp);
  gemm1024_bias_wmma<<<gGrid, gBlk, 0, stream>>>(Yp, Wo, bo, out);
}